// TPLoss_44538810860111
// MI455X (gfx1250) — compile-verified
//
#include <hip/hip_runtime.h>

// TPLoss on MI455X (gfx1250, wave32).
// Memory-bound row reduction: stream pred(f32)+labels(i32) once (1.07 GB),
// HBM floor ~46us @ 23.3 TB/s. Data path: double-buffered async global->LDS
// (GLOBAL_LOAD_ASYNC_TO_LDS_B128, non-temporal, ASYNCcnt-chained), compute
// overlapped.  Per-element math reduced to: s_pl += p*l (fma), s_p += p,
// s_l += l (int add); then TN = N - S_l - S_p + S_pl.

#define WG      256          // 8 waves of 32
#define NDIM    8192         // row length (from reference)
#define CHUNK   2048         // elements per staged chunk
#define NCHUNK  (NDIM / CHUNK)

__device__ __forceinline__ void async_ld_b128_nt(unsigned int lds_byte_off,
                                                 unsigned long long gaddr) {
    // VDST = per-lane LDS byte address, VADDR = per-lane 64-bit global address.
    // Non-temporal: single-use stream, keep it out of the 192MB L2's hot ways.
    asm volatile("global_load_async_to_lds_b128 %0, %1, off th:TH_LOAD_NT"
                 :: "v"(lds_byte_off), "v"(gaddr)
                 : "memory");
}

__device__ __forceinline__ float fast_sigmoid(float x) {
    // sigmoid(x) = 1 / (1 + 2^(-x*log2(e)))   -> v_exp_f32 + v_rcp_f32
    float e = __builtin_amdgcn_exp2f(-x * 1.4426950408889634f);
    return __builtin_amdgcn_rcpf(1.0f + e);
}

__global__ __launch_bounds__(WG) void tploss_row_kernel(
    const float* __restrict__ pred,
    const int*   __restrict__ labels,
    float*       __restrict__ rowOut)
{
    __shared__ __align__(16) float predS[2][CHUNK];
    __shared__ __align__(16) int   labS [2][CHUNK];
    __shared__ float redPL[WG / 32];
    __shared__ float redP [WG / 32];
    __shared__ int   redL [WG / 32];

    const int row = blockIdx.x;
    const int t   = threadIdx.x;

    const float* gp = pred   + (size_t)row * NDIM;
    const int*   gl = labels + (size_t)row * NDIM;

    // ---- stage chunk 0 (4 async b128 per wave: 2 pred + 2 labels) ----
#pragma unroll
    for (int i = 0; i < 2; ++i) {
        const int e = (i * WG + t) * 4;                 // float4-granular element idx
        async_ld_b128_nt((unsigned int)(size_t)&predS[0][e],
                         (unsigned long long)(size_t)(gp + e));
        async_ld_b128_nt((unsigned int)(size_t)&labS[0][e],
                         (unsigned long long)(size_t)(gl + e));
    }

    float s_pl = 0.0f;   // sum p*l
    float s_p  = 0.0f;   // sum p
    int   s_l  = 0;      // sum l (exact integer)

    for (int c = 0; c < NCHUNK; ++c) {
        const int cur = c & 1;

        if (c + 1 < NCHUNK) {
            // prefetch next chunk into the other buffer
            const int nxt  = cur ^ 1;
            const int base = (c + 1) * CHUNK;
#pragma unroll
            for (int i = 0; i < 2; ++i) {
                const int e = (i * WG + t) * 4;
                async_ld_b128_nt((unsigned int)(size_t)&predS[nxt][e],
                                 (unsigned long long)(size_t)(gp + base + e));
                async_ld_b128_nt((unsigned int)(size_t)&labS[nxt][e],
                                 (unsigned long long)(size_t)(gl + base + e));
            }
            // 4 newest (next chunk) may remain outstanding; current chunk done.
            asm volatile("s_wait_asynccnt 4" ::: "memory");
        } else {
            asm volatile("s_wait_asynccnt 0" ::: "memory");
        }

        // Each lane consumes exactly the LDS bytes its own async loads wrote,
        // so per-wave ASYNCcnt ordering is sufficient (no barrier needed).
#pragma unroll
        for (int i = 0; i < 2; ++i) {
            const int e = (i * WG + t) * 4;
            const float4 p4 = *(const float4*)&predS[cur][e];
            const int4   l4 = *(const int4*)  &labS [cur][e];

            const float p0 = fast_sigmoid(p4.x);
            const float p1 = fast_sigmoid(p4.y);
            const float p2 = fast_sigmoid(p4.z);
            const float p3 = fast_sigmoid(p4.w);

            s_pl += p0 * (float)l4.x + p1 * (float)l4.y
                  + p2 * (float)l4.z + p3 * (float)l4.w;
            s_p  += (p0 + p1) + (p2 + p3);
            s_l  += (l4.x + l4.y) + (l4.z + l4.w);
        }
    }

    // ---- wave32 tree reduction (warpSize == 32 on gfx1250) ----
#pragma unroll
    for (int m = 16; m >= 1; m >>= 1) {
        s_pl += __shfl_xor(s_pl, m, 32);
        s_p  += __shfl_xor(s_p,  m, 32);
        s_l  += __shfl_xor(s_l,  m, 32);
    }
    if ((t & 31) == 0) {
        redPL[t >> 5] = s_pl;
        redP [t >> 5] = s_p;
        redL [t >> 5] = s_l;
    }
    __syncthreads();

    if (t == 0) {
        float SPL = 0.0f, SP = 0.0f;
        int   SL  = 0;
#pragma unroll
        for (int w = 0; w < WG / 32; ++w) {
            SPL += redPL[w]; SP += redP[w]; SL += redL[w];
        }
        const float TP = SPL;
        const float TN = ((float)(NDIM - SL) - SP) + SPL;
        rowOut[row] = TP / (1.0f - TP - TN);
    }
}

__global__ __launch_bounds__(256) void tploss_finalize_kernel(
    const float* __restrict__ rowOut, float* __restrict__ out, int B)
{
    __shared__ float red[8];
    const int t = threadIdx.x;

    float s = 0.0f;
    for (int i = t; i < B; i += 256) s += rowOut[i];

#pragma unroll
    for (int m = 16; m >= 1; m >>= 1) s += __shfl_xor(s, m, 32);
    if ((t & 31) == 0) red[t >> 5] = s;
    __syncthreads();

    if (t == 0) {
        float tot = 0.0f;
#pragma unroll
        for (int w = 0; w < 8; ++w) tot += red[w];
        out[0] = -tot / (float)B;
    }
}

extern "C" void kernel_launch(void* const* d_in, const int* in_sizes, int n_in,
                              void* d_out, int out_size, void* d_ws, size_t ws_size,
                              hipStream_t stream) {
    const float* pred   = (const float*)d_in[0];
    const int*   labels = (const int*)  d_in[1];
    float*       rowOut = (float*)d_ws;              // B floats of scratch
    const int    B      = in_sizes[0] / NDIM;        // 4096

    tploss_row_kernel<<<B, WG, 0, stream>>>(pred, labels, rowOut);
    tploss_finalize_kernel<<<1, 256, 0, stream>>>(rowOut, (float*)d_out, B);
}